// NTXentLoss2_40535901340190
// MI455X (gfx1250) — compile-verified
//
#include <hip/hip_runtime.h>
#include <math.h>

// Problem constants (match reference)
#define BB 4096     // batch (identities)
#define NN 8192     // mega-batch N = B*NPOS
#define DD 128      // projection dim
#define INV_TEMP 2.0f   // 1/TEMP, TEMP=0.5
#define EPSN 1e-8f

typedef float v2f __attribute__((ext_vector_type(2)));
typedef float v8f __attribute__((ext_vector_type(8)));

// ---------------------------------------------------------------------------
// Kernel 1: row L2-normalize. One wave32 per row, float4 per lane (128 = 32*4).
// ---------------------------------------------------------------------------
__global__ __launch_bounds__(256) void ntx_normalize(const float* __restrict__ x,
                                                     float* __restrict__ xn) {
  const int row  = blockIdx.x * 8 + (threadIdx.x >> 5);
  const int lane = threadIdx.x & 31;
  float4 v = ((const float4*)(x + (size_t)row * DD))[lane];
  float ss = v.x * v.x + v.y * v.y + v.z * v.z + v.w * v.w;
#pragma unroll
  for (int m = 16; m >= 1; m >>= 1) ss += __shfl_xor(ss, m, 32);
  const float scale = 1.0f / fmaxf(sqrtf(ss), EPSN);
  float4 o;
  o.x = v.x * scale; o.y = v.y * scale; o.z = v.z * scale; o.w = v.w * scale;
  ((float4*)(xn + (size_t)row * DD))[lane] = o;
}

// ---------------------------------------------------------------------------
// Kernel 2: positives. One wave32 per identity b.
// m = (x[b] + x[B+b])/2 ; pos[2b+v] = cos(x[v*B+b], m) / TEMP
// (reference flattens pos [B,NPOS] C-order -> logits row i uses pos[i], b=i/2)
// ---------------------------------------------------------------------------
__global__ __launch_bounds__(256) void ntx_pos(const float* __restrict__ x,
                                               float* __restrict__ posArr) {
  const int b    = blockIdx.x * 8 + (threadIdx.x >> 5);
  const int lane = threadIdx.x & 31;
  float4 a = ((const float4*)(x + (size_t)b * DD))[lane];
  float4 c = ((const float4*)(x + (size_t)(BB + b) * DD))[lane];
  float4 m;
  m.x = 0.5f * (a.x + c.x); m.y = 0.5f * (a.y + c.y);
  m.z = 0.5f * (a.z + c.z); m.w = 0.5f * (a.w + c.w);
  float num0 = a.x * m.x + a.y * m.y + a.z * m.z + a.w * m.w;
  float num1 = c.x * m.x + c.y * m.y + c.z * m.z + c.w * m.w;
  float na   = a.x * a.x + a.y * a.y + a.z * a.z + a.w * a.w;
  float nc   = c.x * c.x + c.y * c.y + c.z * c.z + c.w * c.w;
  float nm   = m.x * m.x + m.y * m.y + m.z * m.z + m.w * m.w;
#pragma unroll
  for (int s = 16; s >= 1; s >>= 1) {
    num0 += __shfl_xor(num0, s, 32);
    num1 += __shfl_xor(num1, s, 32);
    na   += __shfl_xor(na,   s, 32);
    nc   += __shfl_xor(nc,   s, 32);
    nm   += __shfl_xor(nm,   s, 32);
  }
  if (lane == 0) {
    const float dm = fmaxf(sqrtf(nm), EPSN);
    posArr[2 * b + 0] = num0 / (fmaxf(sqrtf(na), EPSN) * dm) * INV_TEMP;
    posArr[2 * b + 1] = num1 / (fmaxf(sqrtf(nc), EPSN) * dm) * INV_TEMP;
  }
}

// ---------------------------------------------------------------------------
// Kernel 3: fused sim = xn@xn^T (WMMA f32 16x16x4) + masked online logsumexp.
// Block = 256 threads = 8 waves, owns one 16-row strip. Each wave walks 64
// column tiles; A fragments (16 rows x K=128) live in registers for reuse.
//
// WMMA f32 16x16x4 per-lane layout (ISA 7.12.2):
//   A: lane l, reg j -> M = l%16,  K = (l/16)*2 + j   (2 VGPRs, v2f)
//   B: lane l, reg j -> N = l%16,  K = (l/16)*2 + j   (B[k][n] = xn[col][k])
//   C: lane l, reg r -> M = (l/16)*8 + r, N = l%16    (8 VGPRs, v8f)
// ---------------------------------------------------------------------------
__global__ __launch_bounds__(256) void ntx_simlse(const float* __restrict__ xn,
                                                  const float* __restrict__ posArr,
                                                  float* __restrict__ partial) {
  __shared__ float sMax[8][16];
  __shared__ float sSum[8][16];

  const int wave = threadIdx.x >> 5;
  const int lane = threadIdx.x & 31;
  const int hi   = lane >> 4;   // lane half (0/1)
  const int lr   = lane & 15;   // lane within half
  const int rowBase = blockIdx.x * 16;

  // Preload A fragments for all K (32 chunks of K=4): 64 VGPRs, reused 64x.
  v2f afrag[32];
  const float* arow = xn + (size_t)(rowBase + lr) * DD + 2 * hi;
#pragma unroll
  for (int kk = 0; kk < 32; ++kk)
    afrag[kk] = *(const v2f*)(arow + kk * 4);

  // Online-LSE state: this lane covers rows M = hi*8 + r of the strip.
  float mx[8], sm[8];
  int rowmod[8];
#pragma unroll
  for (int r = 0; r < 8; ++r) {
    mx[r] = -3.0e38f;
    sm[r] = 0.0f;
    rowmod[r] = (rowBase + hi * 8 + r) & (BB - 1);
  }

  for (int ct = wave; ct < NN / 16; ct += 8) {
    const int colBase = ct * 16;
    const float* brow = xn + (size_t)(colBase + lr) * DD + 2 * hi;
    v8f acc = {0.f, 0.f, 0.f, 0.f, 0.f, 0.f, 0.f, 0.f};
#pragma unroll
    for (int kk = 0; kk < 32; ++kk) {
      v2f bf = *(const v2f*)(brow + kk * 4);
      acc = __builtin_amdgcn_wmma_f32_16x16x4_f32(
          /*neg_a=*/false, afrag[kk], /*neg_b=*/false, bf,
          /*c_mod=*/(short)0, acc, /*reuse_a=*/false, /*reuse_b=*/false);
    }
    const int colmod = (colBase + lr) & (BB - 1);
#pragma unroll
    for (int r = 0; r < 8; ++r) {
      if (colmod != rowmod[r]) {   // mask: drop cols with col%B == row%B
        const float v  = acc[r] * INV_TEMP;
        const float nm = fmaxf(mx[r], v);
        sm[r] = sm[r] * __expf(mx[r] - nm) + __expf(v - nm);
        mx[r] = nm;
      }
    }
  }

  // Merge the 16 lanes of each half (they hold distinct columns of same rows).
#pragma unroll
  for (int r = 0; r < 8; ++r) {
#pragma unroll
    for (int s = 1; s <= 8; s <<= 1) {
      const float om = __shfl_xor(mx[r], s, 32);
      const float os = __shfl_xor(sm[r], s, 32);
      const float nm = fmaxf(mx[r], om);
      sm[r] = sm[r] * __expf(mx[r] - nm) + os * __expf(om - nm);
      mx[r] = nm;
    }
    if (lr == 0) {
      sMax[wave][hi * 8 + r] = mx[r];
      sSum[wave][hi * 8 + r] = sm[r];
    }
  }
  __syncthreads();

  // Merge the 8 waves (distinct column ranges), fold in pos, reduce strip.
  if (threadIdx.x < 16) {
    const int t = threadIdx.x;
    float M = sMax[0][t], S = sSum[0][t];
#pragma unroll
    for (int w = 1; w < 8; ++w) {
      const float om = sMax[w][t], os = sSum[w][t];
      const float nm = fmaxf(M, om);
      S = S * __expf(M - nm) + os * __expf(om - nm);
      M = nm;
    }
    const float p  = posArr[rowBase + t];
    const float nm = fmaxf(M, p);
    S = S * __expf(M - nm) + __expf(p - nm);
    M = nm;
    float contrib = (M + __logf(S)) - p;   // logsumexp(row) - pos
#pragma unroll
    for (int s = 1; s <= 8; s <<= 1) contrib += __shfl_xor(contrib, s, 32);
    if (t == 0) partial[blockIdx.x] = contrib;
  }
}

// ---------------------------------------------------------------------------
// Kernel 4: deterministic reduction of 512 block partials -> loss / N.
// ---------------------------------------------------------------------------
__global__ __launch_bounds__(256) void ntx_finalize(const float* __restrict__ partial,
                                                    float* __restrict__ out) {
  __shared__ float sred[8];
  const int t = threadIdx.x;
  float v = partial[t] + partial[t + 256];
#pragma unroll
  for (int s = 16; s >= 1; s >>= 1) v += __shfl_xor(v, s, 32);
  if ((t & 31) == 0) sred[t >> 5] = v;
  __syncthreads();
  if (t < 8) {
    float w = sred[t];
#pragma unroll
    for (int s = 4; s >= 1; s >>= 1) w += __shfl_xor(w, s, 32);
    if (t == 0) out[0] = w * (1.0f / (float)NN);
  }
}

// ---------------------------------------------------------------------------
extern "C" void kernel_launch(void* const* d_in, const int* in_sizes, int n_in,
                              void* d_out, int out_size, void* d_ws, size_t ws_size,
                              hipStream_t stream) {
  (void)in_sizes; (void)n_in; (void)out_size; (void)ws_size;
  const float* x = (const float*)d_in[0];           // [N, D] fp32
  float* xn      = (float*)d_ws;                    // [N, D] normalized
  float* posArr  = xn + (size_t)NN * DD;            // [N] permuted pos logits
  float* partial = posArr + NN;                     // [N/16] block partials
  float* out     = (float*)d_out;                   // scalar loss

  ntx_normalize<<<NN / 8, 256, 0, stream>>>(x, xn);
  ntx_pos<<<BB / 8, 256, 0, stream>>>(x, posArr);
  ntx_simlse<<<NN / 16, 256, 0, stream>>>(xn, posArr, partial);
  ntx_finalize<<<1, 256, 0, stream>>>(partial, out);
}